// GAT_2834678415393
// MI455X (gfx1250) — compile-verified
//
#include <hip/hip_runtime.h>
#include <hip/hip_bf16.h>
#include <math.h>

// ---------------------------------------------------------------------------
// GATv2 forward on MI455X (gfx1250, wave32).
//  - GEMMs: V_WMMA_F32_16X16X4_F32 (full fp32), weights staged in LDS via the
//    gfx1250 async global->LDS path (global_load_async_to_lds_b128 +
//    s_wait_asynccnt), wave computes a 16 x (NT*16) strip for A-reuse.
//  - Edge stage: wave-per-edge gathers (L2-resident), float atomics for
//    segment max / sum / weighted scatter-add.
// ---------------------------------------------------------------------------

typedef float v2f __attribute__((ext_vector_type(2)));
typedef float v8f __attribute__((ext_vector_type(8)));

#define DIM 64          // hidden dim (H=1, dh=64)
#define NEG_SLOPE 0.2f

// ---------------------------------------------------------------------------
// fp32 WMMA GEMM:  Y[M x N] = X[M x K] @ W[K x N] (+ bias),  N = NT*16
// Block = 8 waves; each wave owns a 16-row M-tile and all NT column tiles.
// W is async-copied to LDS once per block.
//
// ISA 7.12.2 fp32 16x16x4 layouts:
//   A 16x4 : lane = M row (0-15 / 16-31 mirrored); per lane a={X[m,ka],X[m,ka+1]},
//            ka = k + 2*(lane>>4)
//   B  4x16: lane holds column (lane&15); b={W[ka,c], W[ka+1,c]}
//   C 16x16: VGPR v -> row v + 8*(lane>>4), col = lane&15
// ---------------------------------------------------------------------------
template <int NT, int K>
__global__ void gat_wmma_gemm_f32(const float* __restrict__ X,
                                  const float* __restrict__ W,
                                  const float* __restrict__ bias,
                                  float* __restrict__ Y,
                                  int M) {
    constexpr int N = NT * 16;
    __shared__ float sW[128 * 64];               // 32 KB max (K=128, N=64)

    // ---- async copy W (K*N*4 bytes, multiple of 4096) into LDS ----
    {
        const unsigned bytes   = (unsigned)(K * N * 4);
        const unsigned ldsBase = (unsigned)(size_t)(&sW[0]);
        const unsigned long long gBase = (unsigned long long)(size_t)W;
        for (unsigned ofs = (unsigned)threadIdx.x * 16u; ofs < bytes;
             ofs += (unsigned)blockDim.x * 16u) {
            unsigned           ldsa = ldsBase + ofs;
            unsigned long long ga   = gBase + ofs;
            asm volatile("global_load_async_to_lds_b128 %0, %1, off"
                         :: "v"(ldsa), "v"(ga) : "memory");
        }
        asm volatile("s_wait_asynccnt 0x0" ::: "memory");
    }
    __syncthreads();

    const int wave = (int)(threadIdx.x >> 5);
    const int lane = (int)(threadIdx.x & 31);
    const int m0   = ((int)blockIdx.x * 8 + wave) << 4;
    if (m0 < M) {                                // wave-uniform guard, no more barriers
        const int half = lane >> 4;              // 0: lanes 0-15, 1: lanes 16-31
        const int r    = lane & 15;

        v8f acc[NT];
#pragma unroll
        for (int t = 0; t < NT; ++t) acc[t] = (v8f){};

        const float* xrow = X + (size_t)(m0 + r) * (size_t)K;
#pragma unroll
        for (int k = 0; k < K; k += 4) {
            const int ka = k + 2 * half;
            v2f a;
            a.x = xrow[ka];
            a.y = xrow[ka + 1];
#pragma unroll
            for (int t = 0; t < NT; ++t) {
                v2f b;
                b.x = sW[ka * N + t * 16 + r];
                b.y = sW[(ka + 1) * N + t * 16 + r];
                acc[t] = __builtin_amdgcn_wmma_f32_16x16x4_f32(
                             false, a, false, b, (short)0, acc[t], false, false);
            }
        }

#pragma unroll
        for (int t = 0; t < NT; ++t) {
            const float bv = bias ? bias[t * 16 + r] : 0.0f;
#pragma unroll
            for (int v = 0; v < 8; ++v) {
                const int row = v + 8 * half;
                Y[(size_t)(m0 + row) * N + t * 16 + r] = acc[t][v] + bv;
            }
        }
    }
}

// ---------------------------------------------------------------------------
// init: smax = -inf, denom = 0 (over N nodes); agg = 0 (over N*64)
// ---------------------------------------------------------------------------
__global__ void gat_init(float* __restrict__ smax, float* __restrict__ denom,
                         float* __restrict__ agg, int n_nodes) {
    const int i = (int)(blockIdx.x * blockDim.x + threadIdx.x);
    if (i < n_nodes) {
        smax[i]  = -INFINITY;
        denom[i] = 0.0f;
    }
    if (i < n_nodes * DIM) agg[i] = 0.0f;
}

// int-punned float atomic max (works for mixed signs; smax init = -inf)
__device__ __forceinline__ void atomicMaxF(float* addr, float val) {
    if (val >= 0.0f)
        atomicMax((int*)addr, __float_as_int(val));
    else
        atomicMin((unsigned int*)addr, (unsigned int)__float_as_int(val));
}

// ---------------------------------------------------------------------------
// per-edge attention logit: score[e] = sum_d att[d]*leaky_relu(xl[src]+xr[dst])
// one wave per edge; shuffle reduction; atomic float max into smax[dst]
// ---------------------------------------------------------------------------
__global__ void gat_edge_score(const float* __restrict__ xl,
                               const float* __restrict__ xr,
                               const int* __restrict__ src,
                               const int* __restrict__ dst,
                               const float* __restrict__ att,
                               float* __restrict__ score,
                               float* __restrict__ smax, int E) {
    const int wid  = (int)((blockIdx.x * blockDim.x + threadIdx.x) >> 5);
    const int lane = (int)(threadIdx.x & 31);
    if (wid >= E) return;
    const int s = src[wid];
    const int d = dst[wid];
    const float* xls = xl + (size_t)s * DIM;
    const float* xrd = xr + (size_t)d * DIM;

    float p = 0.0f;
#pragma unroll
    for (int i = lane; i < DIM; i += 32) {
        float v = xls[i] + xrd[i];
        v = (v > 0.0f) ? v : NEG_SLOPE * v;
        p += att[i] * v;
    }
#pragma unroll
    for (int off = 16; off > 0; off >>= 1) p += __shfl_xor(p, off, 32);

    if (lane == 0) {
        score[wid] = p;
        atomicMaxF(&smax[d], p);
    }
}

// ---------------------------------------------------------------------------
// ex = exp(score - smax[dst]); score <- ex; denom[dst] += ex
// ---------------------------------------------------------------------------
__global__ void gat_edge_expsum(const int* __restrict__ dst,
                                float* __restrict__ score,
                                const float* __restrict__ smax,
                                float* __restrict__ denom, int E) {
    const int e = (int)(blockIdx.x * blockDim.x + threadIdx.x);
    if (e >= E) return;
    const int d = dst[e];
    const float ex = __expf(score[e] - smax[d]);
    score[e] = ex;
    atomicAdd(&denom[d], ex);
}

// ---------------------------------------------------------------------------
// out[dst] += (ex/denom[dst]) * xl[src]   (wave per edge, 2 dims per lane)
// ---------------------------------------------------------------------------
__global__ void gat_edge_agg(const float* __restrict__ xl,
                             const int* __restrict__ src,
                             const int* __restrict__ dst,
                             const float* __restrict__ score,
                             const float* __restrict__ denom,
                             float* __restrict__ out, int E) {
    const int wid  = (int)((blockIdx.x * blockDim.x + threadIdx.x) >> 5);
    const int lane = (int)(threadIdx.x & 31);
    if (wid >= E) return;
    const int s = src[wid];
    const int d = dst[wid];
    const float alpha = score[wid] / (denom[d] + 1e-16f);
    const float* xls = xl + (size_t)s * DIM;
    float* od = out + (size_t)d * DIM;
#pragma unroll
    for (int i = lane; i < DIM; i += 32)
        atomicAdd(&od[i], alpha * xls[i]);
}

// ---------------------------------------------------------------------------
// h = h + bias ; optional ELU (alpha=1)
// ---------------------------------------------------------------------------
__global__ void gat_bias_act(float* __restrict__ h, const float* __restrict__ bias,
                             int total, int do_elu) {
    const int i = (int)(blockIdx.x * blockDim.x + threadIdx.x);
    if (i >= total) return;
    float v = h[i] + bias[i & (DIM - 1)];
    if (do_elu) v = (v > 0.0f) ? v : (__expf(v) - 1.0f);
    h[i] = v;
}

// ---------------------------------------------------------------------------
// launcher
// inputs: 0 x[N,128] 1 edge_index[2,E] 2 Wl1 3 Wr1 4 a1 5 b1
//         6 Wl2 7 Wr2 8 a2 9 b2 10 Wlin[64,32] 11 blin
// ---------------------------------------------------------------------------
extern "C" void kernel_launch(void* const* d_in, const int* in_sizes, int n_in,
                              void* d_out, int out_size, void* d_ws, size_t ws_size,
                              hipStream_t stream) {
    const float* x    = (const float*)d_in[0];
    const int*   ei   = (const int*)d_in[1];
    const float* Wl1  = (const float*)d_in[2];
    const float* Wr1  = (const float*)d_in[3];
    const float* a1   = (const float*)d_in[4];
    const float* b1   = (const float*)d_in[5];
    const float* Wl2  = (const float*)d_in[6];
    const float* Wr2  = (const float*)d_in[7];
    const float* a2   = (const float*)d_in[8];
    const float* b2   = (const float*)d_in[9];
    const float* Wlin = (const float*)d_in[10];
    const float* blin = (const float*)d_in[11];

    const int N = in_sizes[0] / 128;     // 100000
    const int E = in_sizes[1] / 2;       // 1600000
    const int* src = ei;
    const int* dst = ei + E;

    // workspace carve-up
    char* ws = (char*)d_ws;
    const size_t featB = (size_t)N * DIM * sizeof(float);   // 25.6 MB
    float* xl    = (float*)(ws);
    float* xr    = (float*)(ws + featB);
    float* h     = (float*)(ws + 2 * featB);                // agg / hidden
    float* score = (float*)(ws + 3 * featB);
    float* smax  = (float*)(ws + 3 * featB + (size_t)E * sizeof(float));
    float* denom = smax + N;

    const int TPB = 256;                                    // 8 waves / block
    const int gemmBlocks     = (N / 16 + 7) / 8;            // 8 M-tiles per block
    const int initBlocks     = (N * DIM + TPB - 1) / TPB;
    const int edgeWaveBlocks = (E * 32 + TPB - 1) / TPB;    // wave per edge
    const int edgeThrBlocks  = (E + TPB - 1) / TPB;

    // ---------------- layer 1 ----------------
    gat_wmma_gemm_f32<4, 128><<<gemmBlocks, TPB, 0, stream>>>(x, Wl1, nullptr, xl, N);
    gat_wmma_gemm_f32<4, 128><<<gemmBlocks, TPB, 0, stream>>>(x, Wr1, nullptr, xr, N);
    gat_init<<<initBlocks, TPB, 0, stream>>>(smax, denom, h, N);
    gat_edge_score<<<edgeWaveBlocks, TPB, 0, stream>>>(xl, xr, src, dst, a1, score, smax, E);
    gat_edge_expsum<<<edgeThrBlocks, TPB, 0, stream>>>(dst, score, smax, denom, E);
    gat_edge_agg<<<edgeWaveBlocks, TPB, 0, stream>>>(xl, src, dst, score, denom, h, E);
    gat_bias_act<<<initBlocks, TPB, 0, stream>>>(h, b1, N * DIM, /*elu=*/1);

    // ---------------- layer 2 ----------------
    gat_wmma_gemm_f32<4, 64><<<gemmBlocks, TPB, 0, stream>>>(h, Wl2, nullptr, xl, N);
    gat_wmma_gemm_f32<4, 64><<<gemmBlocks, TPB, 0, stream>>>(h, Wr2, nullptr, xr, N);
    gat_init<<<initBlocks, TPB, 0, stream>>>(smax, denom, h, N);  // reuse h as accumulator
    gat_edge_score<<<edgeWaveBlocks, TPB, 0, stream>>>(xl, xr, src, dst, a2, score, smax, E);
    gat_edge_expsum<<<edgeThrBlocks, TPB, 0, stream>>>(dst, score, smax, denom, E);
    gat_edge_agg<<<edgeWaveBlocks, TPB, 0, stream>>>(xl, src, dst, score, denom, h, E);
    gat_bias_act<<<initBlocks, TPB, 0, stream>>>(h, b2, N * DIM, /*elu=*/0);

    // ---------------- final linear: d_out = h @ Wlin + blin ----------------
    gat_wmma_gemm_f32<2, 64><<<gemmBlocks, TPB, 0, stream>>>(h, Wlin, blin,
                                                             (float*)d_out, N);
}